// GraphConv_65481071401146
// MI455X (gfx1250) — compile-verified
//
#include <hip/hip_runtime.h>
#include <math.h>

// ---------------------------------------------------------------------------
// GraphConv for MI455X (gfx1250, wave32, WMMA + async LDS DMA).
//   out = (adj * deg_inv[None,:]) @ X @ W^T + b  ==  M @ Z + b,  Z = X @ W^T
// adj[i,j] = (dot(sn_i, sn_j)^2 >= 0.1) && i != j, recomputed tile-by-tile
// with V_WMMA_F32_16X16X4_F32 (f32 end-to-end: thresholding must match the
// f32 reference; adjacency never materialized — would be 256 MB).
// Hot kernel double-buffers Z j-tiles into LDS via
// GLOBAL_LOAD_ASYNC_TO_LDS_B128 (ASYNCcnt), overlapping DMA with WMMA.
// ---------------------------------------------------------------------------

typedef __attribute__((ext_vector_type(2))) float v2f;
typedef __attribute__((ext_vector_type(8))) float v8f;

#define N_NODES 8192
#define D_STATE 16
#define F_DIM   256
#define TILE    16
#define NTILES  (N_NODES / TILE)   // 512
#define THRESH  0.1f
#define ZPAD    264                // LDS row stride (floats): 2*264 % 64 == 16
                                   // -> the two lane-halves hit disjoint banks

__device__ __forceinline__ v8f wmma4(v2f a, v2f b, v8f c) {
  // D = A(16x4) * B(4x16) + C, full f32.
  // (neg_a, A, neg_b, B, c_mod, C, reuse_a, reuse_b)
  return __builtin_amdgcn_wmma_f32_16x16x4_f32(
      false, a, false, b, (short)0, c, false, false);
}

// --- kernel 1: row-normalize states, zero degree counters -------------------
__global__ void k_normalize(const float* __restrict__ states,
                            float* __restrict__ sn,
                            unsigned* __restrict__ deg) {
  int i = blockIdx.x * blockDim.x + threadIdx.x;
  if (i >= N_NODES) return;
  const float* row = states + i * D_STATE;
  float v[D_STATE];
  float ss = 0.f;
#pragma unroll
  for (int k = 0; k < D_STATE; ++k) { v[k] = row[k]; ss += v[k] * v[k]; }
  float inv = 1.0f / (sqrtf(ss) + 1e-12f);
  float* o = sn + i * D_STATE;
#pragma unroll
  for (int k = 0; k < D_STATE; ++k) o[k] = v[k] * inv;
  deg[i] = 0u;
}

// --- kernel 2: Z = X @ W^T (WMMA GEMM, 256 K-depth) -------------------------
// grid = NTILES i-tiles, block = 256 (8 waves); wave w covers fo = [32w,32w+32)
__global__ void __launch_bounds__(256) k_xwt(const float* __restrict__ X,
                                             const float* __restrict__ W,
                                             float* __restrict__ Z) {
  int lane = threadIdx.x & 31;
  int wave = threadIdx.x >> 5;
  int r = lane & 15, h = lane >> 4;
  int i0 = blockIdx.x * TILE;
  int fo0 = wave * 32;
  const float* arow  = X + (size_t)(i0 + r) * F_DIM;
  const float* b0row = W + (size_t)(fo0 + r) * F_DIM;
  const float* b1row = W + (size_t)(fo0 + 16 + r) * F_DIM;
  v8f acc0 = {}, acc1 = {};
  for (int k = 0; k < F_DIM; k += 4) {
    v2f a, b0, b1;
    a.x  = arow[k + 2 * h];  a.y  = arow[k + 2 * h + 1];
    b0.x = b0row[k + 2 * h]; b0.y = b0row[k + 2 * h + 1];
    b1.x = b1row[k + 2 * h]; b1.y = b1row[k + 2 * h + 1];
    acc0 = wmma4(a, b0, acc0);
    acc1 = wmma4(a, b1, acc1);
  }
#pragma unroll
  for (int v = 0; v < 8; ++v) {
    int row = i0 + v + 8 * h;
    Z[(size_t)row * F_DIM + fo0 + r]      = acc0[v];
    Z[(size_t)row * F_DIM + fo0 + 16 + r] = acc1[v];
  }
}

// --- kernel 3: degree counts via WMMA fid tiles -----------------------------
// grid = NTILES (i-tile), block = 128 (4 waves); wave w covers j-slice w.
// Transposed dot tile: A = Sn[j-rows], B = Sn[i-rows] -> lane(h,r) holds
// d[v] = dot(sn[j0+v+8h], sn[i0+r]).
__global__ void __launch_bounds__(128) k_degree(const float* __restrict__ sn,
                                                unsigned* __restrict__ deg) {
  int lane = threadIdx.x & 31;
  int wave = threadIdx.x >> 5;
  int r = lane & 15, h = lane >> 4;
  int i0 = blockIdx.x * TILE;
  const float* sirow = sn + (size_t)(i0 + r) * D_STATE;
  float bi[8];
#pragma unroll
  for (int s = 0; s < 4; ++s) {
    bi[2 * s]     = sirow[4 * s + 2 * h];
    bi[2 * s + 1] = sirow[4 * s + 2 * h + 1];
  }
  unsigned cnt = 0;
  int jt0 = wave * (NTILES / 4), jt1 = jt0 + NTILES / 4;
  for (int jt = jt0; jt < jt1; ++jt) {
    int j0 = jt * TILE;
    const float* sjrow = sn + (size_t)(j0 + r) * D_STATE;
    v8f d = {};
#pragma unroll
    for (int s = 0; s < 4; ++s) {
      v2f a, b;
      a.x = sjrow[4 * s + 2 * h]; a.y = sjrow[4 * s + 2 * h + 1];
      b.x = bi[2 * s];            b.y = bi[2 * s + 1];
      d = wmma4(a, b, d);
    }
#pragma unroll
    for (int v = 0; v < 8; ++v) {
      float dot = d[v];
      int jg = j0 + 8 * h + v;
      if (dot * dot >= THRESH && (i0 + r) != jg) cnt++;
    }
  }
  atomicAdd(&deg[i0 + r], cnt);
}

// --- kernel 4: deg_inv ------------------------------------------------------
__global__ void k_deginv(const unsigned* __restrict__ deg,
                         float* __restrict__ dinv) {
  int i = blockIdx.x * blockDim.x + threadIdx.x;
  if (i < N_NODES) dinv[i] = 1.0f / ((float)deg[i] + 1e-6f);
}

// --- async DMA: one 16x256 f32 Z tile -> padded LDS buffer ------------------
// 32 half-rows (512B each); wave w copies half-rows [8w, 8w+8): one
// global_load_async_to_lds_b128 per lane per half-row (32 lanes x 16B = 512B).
__device__ __forceinline__ void prefetch_tile(const float* __restrict__ Z,
                                              int j0, int wave, int lane,
                                              unsigned lds_base_bytes) {
#pragma unroll
  for (int q = 0; q < 8; ++q) {
    int idx  = wave * 8 + q;       // 0..31
    int ro   = idx >> 1;           // tile row 0..15
    int part = idx & 1;            // half-row
    unsigned loff = lds_base_bytes + (unsigned)ro * (ZPAD * 4u) +
                    (unsigned)part * 512u + (unsigned)lane * 16u;
    const float* g = Z + (size_t)(j0 + ro) * F_DIM + part * 128 + lane * 4;
    asm volatile("global_load_async_to_lds_b128 %0, %1, off"
                 :: "v"(loff), "v"(g)
                 : "memory");
  }
}

// --- kernel 5: out = M @ Z + b  (the 34-GFLOP masked aggregation) ----------
// grid = NTILES (i-tile), block = 128 (4 waves); wave w covers f = [64w,64w+64)
// Per j-tile: 4 WMMA (fid, transposed) -> mask*deg_inv -> 8x shfl_xor(16)
// half-swap to re-layout the mask tile as the A operand -> 16 WMMA against
// LDS-staged Z (double-buffered async DMA overlaps the next tile's copy).
__global__ void __launch_bounds__(128) k_agg(const float* __restrict__ sn,
                                             const float* __restrict__ Z,
                                             const float* __restrict__ dinv,
                                             const float* __restrict__ bias,
                                             float* __restrict__ out) {
  __shared__ float zbuf[2][16 * ZPAD];   // 2 x 16.9 KB, bank-conflict-free

  int lane = threadIdx.x & 31;
  int wave = threadIdx.x >> 5;
  int r = lane & 15, h = lane >> 4;
  int i0 = blockIdx.x * TILE;
  int fbase = wave * 64;
  unsigned zb_base[2];
  zb_base[0] = (unsigned)(uintptr_t)&zbuf[0][0];
  zb_base[1] = (unsigned)(uintptr_t)&zbuf[1][0];

  const float* sirow = sn + (size_t)(i0 + r) * D_STATE;
  float bi[8];
#pragma unroll
  for (int s = 0; s < 4; ++s) {
    bi[2 * s]     = sirow[4 * s + 2 * h];
    bi[2 * s + 1] = sirow[4 * s + 2 * h + 1];
  }

  v8f acc[4];
#pragma unroll
  for (int fb = 0; fb < 4; ++fb) {
    float bv = bias[fbase + fb * 16 + r];   // bias depends only on f-column
#pragma unroll
    for (int v = 0; v < 8; ++v) acc[fb][v] = bv;
  }

  prefetch_tile(Z, 0, wave, lane, zb_base[0]);

  for (int jt = 0; jt < NTILES; ++jt) {
    int j0 = jt * TILE;
    // fid tile (transposed): d[v] = dot(sn[j0+v+8h], sn[i0+r])  (global sn)
    const float* sjrow = sn + (size_t)(j0 + r) * D_STATE;
    v8f d = {};
#pragma unroll
    for (int s = 0; s < 4; ++s) {
      v2f a, b;
      a.x = sjrow[4 * s + 2 * h]; a.y = sjrow[4 * s + 2 * h + 1];
      b.x = bi[2 * s];            b.y = bi[2 * s + 1];
      d = wmma4(a, b, d);
    }
    // mask * deg_inv[j], kill self-loops
    const float* dvp = dinv + j0 + 8 * h;
    float me[8], fe[8];
#pragma unroll
    for (int v = 0; v < 8; ++v) {
      float dot = d[v];
      int jg = j0 + 8 * h + v;
      me[v] = (dot * dot >= THRESH && (i0 + r) != jg) ? dvp[v] : 0.0f;
    }
#pragma unroll
    for (int v = 0; v < 8; ++v) fe[v] = __shfl_xor(me[v], 16, 32);
    // Re-layout M[il][jl] into A-operand chunks a2[kk] = {M[r][4kk+2h+0,1]}
    v2f a2[4];
    a2[0].x = h ? fe[2] : me[0]; a2[0].y = h ? fe[3] : me[1];
    a2[1].x = h ? fe[6] : me[4]; a2[1].y = h ? fe[7] : me[5];
    a2[2].x = h ? me[2] : fe[0]; a2[2].y = h ? me[3] : fe[1];
    a2[3].x = h ? me[6] : fe[4]; a2[3].y = h ? me[7] : fe[5];

    // my async copies for zbuf[jt&1] have landed; barrier makes all waves'
    // copies visible AND guarantees zbuf[(jt+1)&1] is fully consumed.
    asm volatile("s_wait_asynccnt 0x0" ::: "memory");
    __syncthreads();
    if (jt + 1 < NTILES)
      prefetch_tile(Z, (jt + 1) * TILE, wave, lane, zb_base[(jt + 1) & 1]);

    // acc[fb] += M(16x16) @ Z_lds[j-tile, f-block]  (K=16 in 4 WMMA steps)
    const float* zb = &zbuf[jt & 1][0];
#pragma unroll
    for (int fb = 0; fb < 4; ++fb) {
      int f0 = fbase + fb * 16;
#pragma unroll
      for (int kk = 0; kk < 4; ++kk) {
        const float* zr = zb + (4 * kk + 2 * h) * ZPAD + f0 + r;
        v2f b2; b2.x = zr[0]; b2.y = zr[ZPAD];
        acc[fb] = wmma4(a2[kk], b2, acc[fb]);
      }
    }
  }

#pragma unroll
  for (int fb = 0; fb < 4; ++fb) {
    int f0 = fbase + fb * 16;
#pragma unroll
    for (int v = 0; v < 8; ++v)
      out[(size_t)(i0 + v + 8 * h) * F_DIM + f0 + r] = acc[fb][v];
  }
}

// ---------------------------------------------------------------------------
extern "C" void kernel_launch(void* const* d_in, const int* in_sizes, int n_in,
                              void* d_out, int out_size, void* d_ws,
                              size_t ws_size, hipStream_t stream) {
  const float* X      = (const float*)d_in[0];  // node_features [8192,256]
  const float* states = (const float*)d_in[1];  // [8192,16]
  const float* W      = (const float*)d_in[2];  // [256,256]
  const float* b      = (const float*)d_in[3];  // [256]
  float* out = (float*)d_out;                   // [8192,256]

  // workspace layout (~8.96 MB total)
  char* ws = (char*)d_ws;
  float*    sn   = (float*)ws;                             // 512 KB
  unsigned* deg  = (unsigned*)(ws + 512 * 1024);           // 32 KB
  float*    dinv = (float*)(ws + 512 * 1024 + 32 * 1024);  // 32 KB
  float*    Z    = (float*)(ws + 576 * 1024);              // 8 MB

  k_normalize<<<N_NODES / 256, 256, 0, stream>>>(states, sn, deg);
  k_xwt<<<NTILES, 256, 0, stream>>>(X, W, Z);
  k_degree<<<NTILES, 128, 0, stream>>>(sn, deg);
  k_deginv<<<N_NODES / 256, 256, 0, stream>>>(deg, dinv);
  k_agg<<<NTILES, 128, 0, stream>>>(sn, Z, dinv, b, out);
}